// BSpline_57286273794764
// MI455X (gfx1250) — compile-verified
//
#include <hip/hip_runtime.h>

typedef unsigned int u32;
typedef unsigned long long u64;
typedef __attribute__((ext_vector_type(4))) unsigned int v4u;
typedef __attribute__((ext_vector_type(8))) int v8i;
typedef __attribute__((ext_vector_type(4))) int v4i;

#define COEF_N   69   // valid coefficients (indices 0..65 actually used)
#define COEF_PAD 72   // LDS tile: 3 OOB elements DMA'd as zero
#define NQUAD    63   // j in [0,62]; q[j] = coeffs[j..j+3]

// Closed-form uniform cubic B-spline weights at local t, quad gathered as
// one 16B-aligned ds_load_b128 from the LDS quad table.
__device__ __forceinline__ float eval_bspline(float xv, const float4* q) {
    float u  = fmaf(xv, 31.5f, 31.5f);          // (x + 1) * 63/2
    float fj = floorf(u);
    fj = fminf(fmaxf(fj, 0.0f), 62.0f);
    float t  = u - fj;
    int   j  = (int)fj;
    float t2 = t * t;
    float t3 = t2 * t;
    float w3 = t3 * (1.0f / 6.0f);                                  // t^3/6
    float w0 = fmaf(t2, 0.5f, fmaf(t, -0.5f, 1.0f / 6.0f)) - w3;    // (1-t)^3/6
    float w1 = fmaf(t3, 0.5f, fmaf(t2, -1.0f, 2.0f / 3.0f));        // (3t^3-6t^2+4)/6
    float w2 = 1.0f - w0 - w1 - w3;                                 // partition of unity
    float4 c = q[j];                             // single ds_load_b128
    float r  = c.x * w0;
    r = fmaf(c.y, w1, r);
    r = fmaf(c.z, w2, r);
    r = fmaf(c.w, w3, r);
    return r;
}

__global__ __launch_bounds__(256) void bspline_eval_kernel(
    const float* __restrict__ x,
    const float* __restrict__ coeffs,
    float* __restrict__ out,
    int n, int n4)
{
    __shared__ float  sc[COEF_PAD];
    __shared__ float4 q[64];                     // 63 quads, padded to 64

    // --- Stage coefficient table into LDS via the Tensor Data Mover. ---
    // TDM is a wave-level DMA op (EXEC ignored); issue from wave 0 only.
    if (threadIdx.x < 32u) {
        u64 ga  = (u64)(uintptr_t)coeffs;
        u32 lds = (u32)(uintptr_t)&sc[0];      // low 32 bits = LDS byte offset

        // D# group 0 (128b): count=1, lds_addr, global_addr[56:0], type=2
        v4u g0;
        g0[0] = 1u;                                           // count=1, is_restore=0
        g0[1] = lds;                                          // lds_addr
        g0[2] = (u32)(ga & 0xFFFFFFFFu);                      // global_addr[31:0]
        g0[3] = (u32)((ga >> 32) & 0x1FFFFFFu) | (2u << 30);  // addr[56:32] | type=2

        // D# group 1 (256b): data_size=4B; tensor_dim0=69 (OOB->0);
        // tile_dim0=72; tensor_dim1=1; dim0_stride=69; no multicast/pad/iterate.
        v8i g1;
        g1[0] = (int)(2u << 16);                 // workgroup_mask=0, data_size=2 (4B)
        g1[1] = (int)((u32)COEF_N << 16);        // tensor_dim0[15:0] (=69)
        g1[2] = (int)(1u << 16);                 // tensor_dim0[31:16]=0, tensor_dim1[15:0]=1
        g1[3] = (int)((u32)COEF_PAD << 16);      // tensor_dim1[31:16]=0, tile_dim0=72
        g1[4] = 0;                               // tile_dim1=0 (unused), tile_dim2=0
        g1[5] = COEF_N;                          // tensor_dim0_stride[31:0]
        g1[6] = 0;                               // stride hi bits / dim1_stride lo
        g1[7] = 0;

        v4i gz  = {0, 0, 0, 0};                  // groups 2/3 unused (<=2D tensor)
        v8i gz8 = {0, 0, 0, 0, 0, 0, 0, 0};      // extra group (clang-23 6-arg form)

        __builtin_amdgcn_tensor_load_to_lds(g0, g1, gz, gz, gz8, 0);
        __builtin_amdgcn_s_wait_tensorcnt(0);
    }
    __syncthreads();   // publish sc[] to all 8 waves (also a compiler mem barrier)

    // --- Build 16B-aligned quad table: q[j] = coeffs[j..j+3]. ---
    if (threadIdx.x < (u32)NQUAD) {
        int j = (int)threadIdx.x;
        float4 v;
        v.x = sc[j];
        v.y = sc[j + 1];
        v.z = sc[j + 2];
        v.w = sc[j + 3];
        q[j] = v;
    }
    __syncthreads();

    // --- Streaming evaluation: float4 per lane, grid-stride. ---
    const float4* __restrict__ x4 = (const float4*)x;
    float4* __restrict__ o4       = (float4*)out;
    int tid    = (int)(blockIdx.x * blockDim.x + threadIdx.x);
    int stride = (int)(gridDim.x * blockDim.x);

    for (int i = tid; i < n4; i += stride) {
        float4 v = x4[i];
        float4 r;
        r.x = eval_bspline(v.x, q);
        r.y = eval_bspline(v.y, q);
        r.z = eval_bspline(v.z, q);
        r.w = eval_bspline(v.w, q);
        o4[i] = r;
    }

    // Scalar tail (N here is a multiple of 4, but stay general).
    int tail_base = n4 << 2;
    int tail      = n - tail_base;
    if (tid < tail) {
        out[tail_base + tid] = eval_bspline(x[tail_base + tid], q);
    }
}

extern "C" void kernel_launch(void* const* d_in, const int* in_sizes, int n_in,
                              void* d_out, int out_size, void* d_ws, size_t ws_size,
                              hipStream_t stream) {
    const float* x      = (const float*)d_in[0];   // (N,) float32 in [-1, 1)
    const float* coeffs = (const float*)d_in[1];   // (69,) float32
    // d_in[2] is the extended grid; spacing is uniform (h = 2/63) so it is
    // derived analytically instead of loaded.
    float* out = (float*)d_out;

    int n  = in_sizes[0];
    int n4 = n >> 2;

    const int threads = 256;                       // 8 wave32s per block
    int blocks = (n4 + threads - 1) / threads;
    if (blocks > 4096) blocks = 4096;              // grid-stride beyond this
    if (blocks < 1) blocks = 1;

    bspline_eval_kernel<<<blocks, threads, 0, stream>>>(x, coeffs, out, n, n4);
}